// res_gcn_meta_block_15616501088317
// MI455X (gfx1250) — compile-verified
//
#include <hip/hip_runtime.h>
#include <cmath>

// Problem constants (match reference)
#define BATCH 2
#define NPTS  8192
#define CCH   128
#define KNN   16
#define NK    17          // K+1 (self included, dropped at the end)
#define HID   32
#define OI    (CCH*CCH)
#define FEAT  (BATCH*CCH*NPTS)

typedef __attribute__((ext_vector_type(2))) float v2f;
typedef __attribute__((ext_vector_type(8))) float v8f;

__device__ __forceinline__ float lrelu(float v) { return v > 0.0f ? v : 0.2f * v; }

// ---------------------------------------------------------------------------
// Kernel 1: brute-force KNN (top-17 incl. self), register insertion sort,
// candidate points streamed through LDS in 1024-point tiles.
// grid (N/256, B), block 256
// ---------------------------------------------------------------------------
#define KTILE 1024
__global__ __launch_bounds__(256) void knn_kernel(const float* __restrict__ xyz,
                                                  int* __restrict__ idx_out) {
  __shared__ float sP[KTILE * 3];
  const int b = blockIdx.y;
  const int q = blockIdx.x * 256 + threadIdx.x;
  const float* X = xyz + (size_t)b * NPTS * 3;

  const float qx = X[q * 3 + 0];
  const float qy = X[q * 3 + 1];
  const float qz = X[q * 3 + 2];

  float dist[NK];
  int   ind[NK];
#pragma unroll
  for (int j = 0; j < NK; ++j) { dist[j] = 3.4e38f; ind[j] = 0; }

  for (int t0 = 0; t0 < NPTS; t0 += KTILE) {
    __syncthreads();
    for (int e = threadIdx.x; e < KTILE * 3; e += 256)
      sP[e] = X[(size_t)t0 * 3 + e];
    __syncthreads();
    for (int c = 0; c < KTILE; ++c) {
      const float dx = qx - sP[c * 3 + 0];
      const float dy = qy - sP[c * 3 + 1];
      const float dz = qz - sP[c * 3 + 2];
      const float d  = dx * dx + dy * dy + dz * dz;
      if (d < dist[NK - 1]) {
        // branchless sorted insert (ascending), fully unrolled (static indexing)
        float dd = d; int ii = t0 + c;
#pragma unroll
        for (int j = 0; j < NK; ++j) {
          const bool  sm = dd < dist[j];
          const float td = dist[j];
          const int   ti = ind[j];
          dist[j] = sm ? dd : td;
          ind[j]  = sm ? ii : ti;
          dd = sm ? td : dd;
          ii = sm ? ti : ii;
        }
      }
    }
  }
  int* o = idx_out + ((size_t)b * NPTS + q) * KNN;
#pragma unroll
  for (int j = 1; j < NK; ++j) o[j - 1] = ind[j];   // drop self (slot 0)
}

// ---------------------------------------------------------------------------
// Kernel 2: scalar-MLP -> 128x128 weight matrix
// W[o] = b12[o] + s*skip_w[o] + skip_b[o] + sum_j w12[o,j]*lrelu(s*w11[j]+b11[j])
// ---------------------------------------------------------------------------
__global__ __launch_bounds__(256) void mlp_kernel(const float* __restrict__ w11,
                                                  const float* __restrict__ b11,
                                                  const float* __restrict__ w12,
                                                  const float* __restrict__ b12,
                                                  const float* __restrict__ sw,
                                                  const float* __restrict__ sb,
                                                  float* __restrict__ W, float s) {
  __shared__ float h[HID];
  if (threadIdx.x < HID) h[threadIdx.x] = lrelu(s * w11[threadIdx.x] + b11[threadIdx.x]);
  __syncthreads();
  for (int o = blockIdx.x * 256 + threadIdx.x; o < OI; o += gridDim.x * 256) {
    float acc = b12[o] + s * sw[o] + sb[o];
#pragma unroll
    for (int j = 0; j < HID; ++j) acc += w12[o * HID + j] * h[j];
    W[o] = acc;
  }
}

// ---------------------------------------------------------------------------
// Kernel 3: reg = 0.5 * (||W1||^2 + ||W2||^2)   (W1,W2 contiguous in ws)
// ---------------------------------------------------------------------------
__global__ __launch_bounds__(256) void reg_kernel(const float* __restrict__ W12,
                                                  float* __restrict__ out_reg) {
  __shared__ float red[256];
  float acc = 0.0f;
  for (int i = threadIdx.x; i < 2 * OI; i += 256) { const float w = W12[i]; acc += w * w; }
  red[threadIdx.x] = acc;
  __syncthreads();
  for (int s = 128; s > 0; s >>= 1) {
    if (threadIdx.x < s) red[threadIdx.x] += red[threadIdx.x + s];
    __syncthreads();
  }
  if (threadIdx.x == 0) *out_reg = 0.5f * red[0];
}

// ---------------------------------------------------------------------------
// Kernel 4: G[b,c,n] = sum_k lrelu(P[b,c,idx[b,n,k]])
// One block per (b,c) row; lrelu'd row cached in LDS (32 KB), 16 ds gathers/n.
// ---------------------------------------------------------------------------
__global__ __launch_bounds__(256) void gather_kernel(const float* __restrict__ points,
                                                     const int* __restrict__ idx,
                                                     float* __restrict__ G) {
  __shared__ float row[NPTS];
  const int bc = blockIdx.x;          // b*128 + c
  const int b  = bc >> 7;
  const float* src = points + (size_t)bc * NPTS;
  for (int i = threadIdx.x; i < NPTS; i += 256) row[i] = lrelu(src[i]);
  __syncthreads();
  const int* ib = idx + (size_t)b * NPTS * KNN;
  float* g = G + (size_t)bc * NPTS;
  for (int n = threadIdx.x; n < NPTS; n += 256) {
    const int* ip = ib + (size_t)n * KNN;
    float sum = 0.0f;
#pragma unroll
    for (int k = 0; k < KNN; ++k) sum += row[ip[k]];
    g[n] = sum;
  }
}

// ---------------------------------------------------------------------------
// Kernel 5: fused WMMA GEMM + epilogue
//   features = (W1 @ lrelu(P) + W2 @ G) / 17 + P
// Block = 8 waves; wave w owns output rows [16w,16w+16) for a 16-col n-tile.
// K-loop: 32 steps of V_WMMA_F32_16X16X4_F32 x2 into a shared f32 accumulator.
// fp32 A 16x4 fragment layout (ISA 7.12.2): lane m=lane&15 holds M=m;
// VGPR0 = K {0|2}, VGPR1 = K {1|3} selected by lane half.
// ---------------------------------------------------------------------------
__global__ __launch_bounds__(256) void fused_out_kernel(const float* __restrict__ points,
                                                        const float* __restrict__ W1,
                                                        const float* __restrict__ W2,
                                                        const float* __restrict__ G,
                                                        float* __restrict__ out) {
  __shared__ float Pt[CCH * 16];   // lrelu(P) tile, row-major [i][n]
  __shared__ float Gt[CCH * 16];   // G tile
  const int b  = blockIdx.y;
  const int n0 = blockIdx.x * 16;
  const float* P  = points + (size_t)b * CCH * NPTS;
  const float* Gb = G      + (size_t)b * CCH * NPTS;

  for (int e = threadIdx.x; e < CCH * 16; e += 256) {
    const int i = e >> 4, n = e & 15;
    Pt[e] = lrelu(P[(size_t)i * NPTS + n0 + n]);
    Gt[e] = Gb[(size_t)i * NPTS + n0 + n];
  }
  __syncthreads();

  const int wave = threadIdx.x >> 5;      // wave32
  const int lane = threadIdx.x & 31;
  const int m    = lane & 15;
  const int hh   = lane >> 4;             // lane half
  const int h2   = hh * 2;
  const int o0   = wave * 16;

  const float* w1row = W1 + (size_t)(o0 + m) * CCH;
  const float* w2row = W2 + (size_t)(o0 + m) * CCH;

  v8f acc = {};
#pragma unroll 4
  for (int kb = 0; kb < CCH; kb += 4) {
    const int ka = kb + h2;
    // A fragments (16x4 fp32) straight from L2-resident W matrices
    v2f a1 = { w1row[ka], w1row[ka + 1] };
    v2f bp = { Pt[ka * 16 + m], Pt[(ka + 1) * 16 + m] };
    acc = __builtin_amdgcn_wmma_f32_16x16x4_f32(false, a1, false, bp,
                                                (short)0, acc, false, false);
    v2f a2 = { w2row[ka], w2row[ka + 1] };
    v2f bg = { Gt[ka * 16 + m], Gt[(ka + 1) * 16 + m] };
    acc = __builtin_amdgcn_wmma_f32_16x16x4_f32(false, a2, false, bg,
                                                (short)0, acc, false, false);
  }

  // D layout: VGPR r, lanes<16 -> M=r, lanes>=16 -> M=r+8; N = lane&15
  const float inv = 1.0f / (float)NK;
#pragma unroll
  for (int r = 0; r < 8; ++r) {
    const int row = o0 + r + 8 * hh;
    const size_t off = (size_t)b * CCH * NPTS + (size_t)row * NPTS + n0 + m;
    out[off] = acc[r] * inv + points[off];
  }
}

// ---------------------------------------------------------------------------
// Launch: out layout = [features (2*128*8192 f32)] [reg (1 f32)] [idx (2*8192*16 i32 bits)]
// ws layout = [W1 16384 f32][W2 16384 f32][G 2*128*8192 f32]  (~8.5 MB)
// ---------------------------------------------------------------------------
extern "C" void kernel_launch(void* const* d_in, const int* in_sizes, int n_in,
                              void* d_out, int out_size, void* d_ws, size_t ws_size,
                              hipStream_t stream) {
  const float* points = (const float*)d_in[0];
  const float* xyz    = (const float*)d_in[1];
  const float* m1_w11 = (const float*)d_in[2];
  const float* m1_b11 = (const float*)d_in[3];
  const float* m1_w12 = (const float*)d_in[4];
  const float* m1_b12 = (const float*)d_in[5];
  const float* m1_sw  = (const float*)d_in[6];
  const float* m1_sb  = (const float*)d_in[7];
  const float* m2_w11 = (const float*)d_in[8];
  const float* m2_b11 = (const float*)d_in[9];
  const float* m2_w12 = (const float*)d_in[10];
  const float* m2_b12 = (const float*)d_in[11];
  const float* m2_sw  = (const float*)d_in[12];
  const float* m2_sb  = (const float*)d_in[13];
  (void)d_in[14]; (void)in_sizes; (void)n_in; (void)out_size; (void)ws_size;

  float* out  = (float*)d_out;
  float* regp = out + FEAT;
  int*   idxp = (int*)(out + FEAT + 1);

  float* W1 = (float*)d_ws;
  float* W2 = W1 + OI;
  float* G  = W2 + OI;

  // s = sqrt(arange(1.1,9.1,0.1)[28]) / sqrt(arange[...][-1]); idx 28 since
  // int((2.0**2 - 1.1)/0.1) == 28 in IEEE double.
  const double u28 = 1.1 + 0.1 * 28.0;
  const double u79 = 1.1 + 0.1 * 79.0;
  const float  s   = (float)(sqrt(u28) / sqrt(u79));

  knn_kernel<<<dim3(NPTS / 256, BATCH), 256, 0, stream>>>(xyz, idxp);
  mlp_kernel<<<16, 256, 0, stream>>>(m1_w11, m1_b11, m1_w12, m1_b12, m1_sw, m1_sb, W1, s);
  mlp_kernel<<<16, 256, 0, stream>>>(m2_w11, m2_b11, m2_w12, m2_b12, m2_sw, m2_sb, W2, s);
  reg_kernel<<<1, 256, 0, stream>>>(W1, regp);
  gather_kernel<<<BATCH * CCH, 256, 0, stream>>>(points, idxp, G);
  fused_out_kernel<<<dim3(NPTS / 16, BATCH), 256, 0, stream>>>(points, W1, W2, G, out);
}